// GaussianMoments_87960930222854
// MI455X (gfx1250) — compile-verified
//
#include <hip/hip_runtime.h>
#include <hip/hip_bf16.h>

typedef __attribute__((ext_vector_type(2))) float v2f;
typedef __attribute__((ext_vector_type(8))) float v8f;

// ---------------------------------------------------------------------------
// WMMA helper: D(16x16,f32) = A(16x4,f32) * B(4x16,f32) + C, full f32 precision
// ---------------------------------------------------------------------------
__device__ __forceinline__ v8f wmma_k4(v2f a, v2f b, v8f c) {
  return __builtin_amdgcn_wmma_f32_16x16x4_f32(
      /*neg_a=*/false, a, /*neg_b=*/false, b,
      /*c_mod=*/(short)0, c, /*reuse_a=*/false, /*reuse_b=*/false);
}

// Build a 2-VGPR fragment for a K-chunk: lanes 0-15 take K = 4c,4c+1,
// lanes 16-31 take K = 4c+2,4c+3 (identical layout for A rows and B columns
// of a Gram/product since lane = row(A) = col(B)).
__device__ __forceinline__ v2f make_frag(float x0, float x1, float x2, float x3, bool g) {
  v2f r;
  r.x = g ? x2 : x0;
  r.y = g ? x3 : x1;
  return r;
}
#define FRAG4(ARR, K, c, g) make_frag(                         \
    ((4*(c)+0) < (K)) ? (ARR)[4*(c)+0] : 0.0f,                 \
    ((4*(c)+1) < (K)) ? (ARR)[4*(c)+1] : 0.0f,                 \
    ((4*(c)+2) < (K)) ? (ARR)[4*(c)+2] : 0.0f,                 \
    ((4*(c)+3) < (K)) ? (ARR)[4*(c)+3] : 0.0f, (g))

// Store upper-triangular (r<=s) 7x7 block; lane s holds column s in d[r].
__device__ __forceinline__ void store_triu2(float* base, v8f d, int lane, int stride) {
  const int P2R[7] = {0, 6, 11, 15, 18, 20, 21};  // p2(r,s) = P2R[r] + s
#pragma unroll
  for (int r = 0; r < 7; ++r)
    if (lane >= r && lane < 7)
      base[(P2R[r] + lane) * stride] = d[r];
}

// Store triu3 (r<=s<=t) entries of c4 for this fixed t.
__device__ __forceinline__ void store_triu3(float* base, v8f d, int lane, int t) {
  const int OFFI[7] = {0, 28, 49, 64, 74, 80, 83};
#pragma unroll
  for (int r = 0; r < 7; ++r)
    if (lane >= r && lane <= t) {
      int s = lane;
      int idx = OFFI[r] + 7 * (s - r) - ((s - 1 + r) * (s - r)) / 2 + (t - s);
      base[idx] = d[r];
    }
}

// ---------------------------------------------------------------------------
// Kernel 0: zero the per-atom accumulator (nAtoms x 112 f32) in workspace
// ---------------------------------------------------------------------------
__global__ void gm_zero_kernel(float* __restrict__ p, int n) {
  int i = blockIdx.x * blockDim.x + threadIdx.x;
  if (i < n) p[i] = 0.0f;
}

// ---------------------------------------------------------------------------
// Kernel 1: edge phase. acc[atom][r][c] += e_r * G_c  (112 atomics per edge,
// accumulator is 22.4MB -> L2 resident on the 192MB L2).
// ---------------------------------------------------------------------------
__global__ __launch_bounds__(256) void gm_edge_kernel(
    const float* __restrict__ pos, const float* __restrict__ shifts,
    const float* __restrict__ emb, const int* __restrict__ species,
    const int* __restrict__ eidx, float* __restrict__ acc, int nEdges) {
  int e = blockIdx.x * blockDim.x + threadIdx.x;
  if (e >= nEdges) return;
  int i = eidx[e];
  int j = eidx[nEdges + e];
  float rx = pos[3 * i + 0] - pos[3 * j + 0] - shifts[3 * e + 0];
  float ry = pos[3 * i + 1] - pos[3 * j + 1] - shifts[3 * e + 1];
  float rz = pos[3 * i + 2] - pos[3 * j + 2] - shifts[3 * e + 2];
  float dd = sqrtf(rx * rx + ry * ry + rz * rz);
  if (dd >= 5.0f) return;  // cutoff -> zero contribution, skip atomics
  float inv = 1.0f / (dd + 1e-12f);
  float vx = rx * inv, vy = ry * inv, vz = rz * inv;
  float cut = 0.5f * (cosf(0.6283185307179586f * dd) + 1.0f);  // pi/R_CUTOFF

  float basis[7];
#pragma unroll
  for (int b = 0; b < 7; ++b) {
    float cen = 0.5f + 0.6428571428571429f * (float)b;
    float tt = dd - cen;
    basis[b] = 1.0569004f * expf(-1.96f * tt * tt);  // rad_norm, betta
  }
  int si = species[i], sj = species[j];
  const float* __restrict__ W = emb + (size_t)(si * 119 + sj) * 49;

  float G4 = vx * vx, G5 = vx * vy, G6 = vx * vz, G7 = vy * vy, G8 = vy * vz;
  float G[16] = {1.0f, vx, vy, vz, G4, G5, G6, G7, G8,
                 G4 * vx, G4 * vy, G4 * vz, vx * G7, G5 * vz, G7 * vy, G7 * vz};

  float* __restrict__ arow = acc + (size_t)i * 112;
  const float cutn = cut * 0.3779644730092272f;  // 1/sqrt(7)
#pragma unroll
  for (int r = 0; r < 7; ++r) {
    float er = 0.0f;
#pragma unroll
    for (int b = 0; b < 7; ++b) er += basis[b] * W[r * 7 + b];
    er *= cutn;
#pragma unroll
    for (int c = 0; c < 16; ++c) atomicAdd(&arow[r * 16 + c], er * G[c]);
  }
}

// ---------------------------------------------------------------------------
// Kernel 2: atom phase. One wave32 per atom. All contractions via f32 WMMA.
// ---------------------------------------------------------------------------
__global__ __launch_bounds__(256) void gm_atom_kernel(
    const float* __restrict__ acc, float* __restrict__ out, int nAtoms) {
  const int lane = threadIdx.x & 31;
  const int atom = blockIdx.x * 8 + (threadIdx.x >> 5);
  if (atom >= nAtoms) return;  // wave-uniform: EXEC stays all-ones for WMMA
  const int row = lane & 15;
  const bool g = lane >= 16;
  const float* __restrict__ A = acc + (size_t)atom * 112;
  float* __restrict__ o = out + (size_t)atom * 910;

  // Per-lane row of M1 (7x3), M2 (7x9 full symmetric), M3 (7x27 full symmetric)
  float M1[3] = {0.f, 0.f, 0.f};
  float M2[9] = {0.f, 0.f, 0.f, 0.f, 0.f, 0.f, 0.f, 0.f, 0.f};
  float M3[27];
#pragma unroll
  for (int q = 0; q < 27; ++q) M3[q] = 0.f;

  if (row < 7) {
    const float* ar = A + row * 16;
    float a0 = ar[0], a1 = ar[1], a2 = ar[2], a3 = ar[3];
    float a4 = ar[4], a5 = ar[5], a6 = ar[6], a7 = ar[7], a8 = ar[8];
    float a9 = ar[9], a10 = ar[10], a11 = ar[11], a12 = ar[12];
    float a13 = ar[13], a14 = ar[14], a15 = ar[15];
    M1[0] = a1; M1[1] = a2; M1[2] = a3;
    float zz = a0 - a4 - a7;
    M2[0] = a4; M2[1] = a5; M2[2] = a6;
    M2[3] = a5; M2[4] = a7; M2[5] = a8;
    M2[6] = a6; M2[7] = a8; M2[8] = zz;
    // 10 unique m3 components: xxx xxy xxz xyy xyz xzz yyy yyz yzz zzz
    float u[10] = {a9, a10, a11, a12, a13, a1 - a9 - a12,
                   a14, a15, a2 - a10 - a14, a3 - a11 - a15};
    const int map[27] = {0, 1, 2, 1, 3, 4, 2, 4, 5,
                         1, 3, 4, 3, 6, 7, 4, 7, 8,
                         2, 4, 5, 4, 7, 8, 5, 8, 9};
#pragma unroll
    for (int q = 0; q < 27; ++q) M3[q] = u[map[q]];
  }

  const v8f z8 = {0.f, 0.f, 0.f, 0.f, 0.f, 0.f, 0.f, 0.f};

  // A fragments (also used as B for Gram products; identical register layout)
  v2f a1f  = FRAG4(M1, 3, 0, g);
  v2f a2f0 = FRAG4(M2, 9, 0, g), a2f1 = FRAG4(M2, 9, 1, g), a2f2 = FRAG4(M2, 9, 2, g);
  v2f a3f0 = FRAG4(M3, 27, 0, g), a3f1 = FRAG4(M3, 27, 1, g), a3f2 = FRAG4(M3, 27, 2, g);
  v2f a3f3 = FRAG4(M3, 27, 3, g), a3f4 = FRAG4(M3, 27, 4, g), a3f5 = FRAG4(M3, 27, 5, g);
  v2f a3f6 = FRAG4(M3, 27, 6, g);

  // c0
  if (lane < 7) o[lane] = A[lane * 16];

  // c1 = Gram(M1): K=3
  v8f d = wmma_k4(a1f, a1f, z8);
  store_triu2(o + 7, d, lane, 1);

  // c2 = Gram(M2): K=9
  d = z8;
  d = wmma_k4(a2f0, a2f0, d);
  d = wmma_k4(a2f1, a2f1, d);
  d = wmma_k4(a2f2, a2f2, d);
  store_triu2(o + 35, d, lane, 1);

  // c3 = Gram(M3): K=27
  d = z8;
  d = wmma_k4(a3f0, a3f0, d); d = wmma_k4(a3f1, a3f1, d);
  d = wmma_k4(a3f2, a3f2, d); d = wmma_k4(a3f3, a3f3, d);
  d = wmma_k4(a3f4, a3f4, d); d = wmma_k4(a3f5, a3f5, d);
  d = wmma_k4(a3f6, a3f6, d);
  store_triu2(o + 63, d, lane, 1);

  for (int t = 0; t < 7; ++t) {
    // broadcast row t of m1/m2 (L0-hot reads, same address all lanes)
    const float* at = A + t * 16;
    float b0 = at[0], b1 = at[1], b2 = at[2], b3 = at[3], b4 = at[4];
    float b5 = at[5], b6 = at[6], b7 = at[7], b8 = at[8];
    float m2t[9] = {b4, b5, b6, b5, b7, b8, b6, b8, b0 - b4 - b7};
    float m1t[3] = {b1, b2, b3};

    // ---- c5[r,s,t] = sum_ij m1[r,i] m1[s,j] m2[t,i,j] ----
    // B[k=i][s] = sum_j m2t[i,j] * m1[s,j]   (K=3)
    float c5b0 = m2t[0] * M1[0] + m2t[1] * M1[1] + m2t[2] * M1[2];
    float c5b1 = m2t[3] * M1[0] + m2t[4] * M1[1] + m2t[5] * M1[2];
    float c5b2 = m2t[6] * M1[0] + m2t[7] * M1[1] + m2t[8] * M1[2];
    d = wmma_k4(a1f, make_frag(c5b0, c5b1, c5b2, 0.f, g), z8);
    store_triu2(o + 175 + t, d, lane, 7);

    // ---- c4[r,s,t] = sum_ijk m2[r,ij] m2[s,ik] m2[t,jk] ----
    // B[(i,j)][s] = (m2[s] @ m2[t]^T)[i,j]   (K=9)
    float E[9];
#pragma unroll
    for (int ij = 0; ij < 9; ++ij) {
      const int ii = ij / 3, jj = ij % 3;
      E[ij] = M2[ii * 3 + 0] * m2t[jj * 3 + 0] +
              M2[ii * 3 + 1] * m2t[jj * 3 + 1] +
              M2[ii * 3 + 2] * m2t[jj * 3 + 2];
    }
    d = z8;
    d = wmma_k4(a2f0, FRAG4(E, 9, 0, g), d);
    d = wmma_k4(a2f1, FRAG4(E, 9, 1, g), d);
    d = wmma_k4(a2f2, FRAG4(E, 9, 2, g), d);
    store_triu3(o + 91, d, lane, t);

    // ---- c6[r,s,t] = sum_ijkl m3[r,ijk] m3[s,ijl] m2[t,kl] ----
    // B[(i,j,k)][s] = sum_l m3[s,i,j,l] * m2t[k,l]   (K=27)
    float F[27];
#pragma unroll
    for (int q = 0; q < 27; ++q) {
      const int ij3 = (q / 3) * 3;
      const int k3 = (q % 3) * 3;
      F[q] = M3[ij3 + 0] * m2t[k3 + 0] + M3[ij3 + 1] * m2t[k3 + 1] +
             M3[ij3 + 2] * m2t[k3 + 2];
    }
    d = z8;
    d = wmma_k4(a3f0, FRAG4(F, 27, 0, g), d);
    d = wmma_k4(a3f1, FRAG4(F, 27, 1, g), d);
    d = wmma_k4(a3f2, FRAG4(F, 27, 2, g), d);
    d = wmma_k4(a3f3, FRAG4(F, 27, 3, g), d);
    d = wmma_k4(a3f4, FRAG4(F, 27, 4, g), d);
    d = wmma_k4(a3f5, FRAG4(F, 27, 5, g), d);
    d = wmma_k4(a3f6, FRAG4(F, 27, 6, g), d);
    store_triu2(o + 371 + t, d, lane, 7);

    // ---- c7[r,s,t] = sum_ijk m3[r,ijk] m2[s,ij] m1[t,k] ----
    // B[(i,j,k)][s] = m2[s,ij] * m1t[k]   (K=27)
    float H[27];
#pragma unroll
    for (int q = 0; q < 27; ++q) H[q] = M2[q / 3] * m1t[q % 3];
    d = z8;
    d = wmma_k4(a3f0, FRAG4(H, 27, 0, g), d);
    d = wmma_k4(a3f1, FRAG4(H, 27, 1, g), d);
    d = wmma_k4(a3f2, FRAG4(H, 27, 2, g), d);
    d = wmma_k4(a3f3, FRAG4(H, 27, 3, g), d);
    d = wmma_k4(a3f4, FRAG4(H, 27, 4, g), d);
    d = wmma_k4(a3f5, FRAG4(H, 27, 5, g), d);
    d = wmma_k4(a3f6, FRAG4(H, 27, 6, g), d);
#pragma unroll
    for (int r = 0; r < 7; ++r)
      if (lane < 7) o[567 + r * 49 + lane * 7 + t] = d[r];
  }
}

extern "C" void kernel_launch(void* const* d_in, const int* in_sizes, int n_in,
                              void* d_out, int out_size, void* d_ws, size_t ws_size,
                              hipStream_t stream) {
  const float* pos    = (const float*)d_in[0];
  const float* shifts = (const float*)d_in[1];
  const float* emb    = (const float*)d_in[2];
  const int* species  = (const int*)d_in[3];
  const int* eidx     = (const int*)d_in[4];
  const int nAtoms = in_sizes[3];
  const int nEdges = in_sizes[4] / 2;

  float* acc = (float*)d_ws;           // nAtoms * 112 floats (22.4 MB)
  const int accN = nAtoms * 112;

  gm_zero_kernel<<<(accN + 255) / 256, 256, 0, stream>>>(acc, accN);
  gm_edge_kernel<<<(nEdges + 255) / 256, 256, 0, stream>>>(
      pos, shifts, emb, species, eidx, acc, nEdges);
  gm_atom_kernel<<<(nAtoms + 7) / 8, 256, 0, stream>>>(acc, (float*)d_out, nAtoms);
}